// GaussianAttention_78262894068500
// MI455X (gfx1250) — compile-verified
//
#include <hip/hip_runtime.h>
#include <cstdint>

typedef float v2f __attribute__((ext_vector_type(2)));
typedef float v8f __attribute__((ext_vector_type(8)));
typedef unsigned int u32x4 __attribute__((ext_vector_type(4)));
typedef int i32x4 __attribute__((ext_vector_type(4)));
typedef int i32x8 __attribute__((ext_vector_type(8)));

#define B_ 8
#define U_ 512
#define T_ 1024
#define H_ 512
#define D_ 128
#define K_ 10

#define TCHUNK 64
#define ROWSTR 144   // 128 data dwords + 16 pad dwords (TDM D# pad feature)

// -----------------------------------------------------------------------------
// TDM: DMA one [TCHUNK x 128] f32 tile of ctx into LDS (padded rows).
// D# group0: count=1, lds_addr, global_addr[56:0], type=2.
// D# group1: data_size=4B, pad_enable, pad_interval=128dw, pad_amount=16dw,
//            tensor_dim0=128, tensor_dim1=1024, tile_dim0=128, tile_dim1=64,
//            tensor_dim0_stride=128.
// -----------------------------------------------------------------------------
__device__ __forceinline__ void tdm_load_ctx_chunk(const float* gsrc,
                                                   unsigned int lds_off) {
  unsigned long long ga = (unsigned long long)(uintptr_t)gsrc;
  u32x4 g0;
  g0.x = 1u;                                   // count=1 (valid user descriptor)
  g0.y = lds_off;                              // lds_addr (bytes)
  g0.z = (unsigned int)ga;                     // global_addr[31:0]
  g0.w = (unsigned int)((ga >> 32) & 0x01FFFFFFu) | (2u << 30); // [56:32] | type=2
  i32x8 g1;
  g1[0] = (int)0x1F920000u;  // data_size=2(4B) | pad_en | interval=6(128dw) | amount=15(16dw)
  g1[1] = (int)(128u << 16);                   // tensor_dim0 = 128 (low 16)
  g1[2] = (int)(1024u << 16);                  // dim0_hi=0 | tensor_dim1=1024 (low 16)
  g1[3] = (int)(128u << 16);                   // dim1_hi=0 | tile_dim0 = 128
  g1[4] = (int)TCHUNK;                         // tile_dim1 = 64, tile_dim2 = 0
  g1[5] = (int)128;                            // tensor_dim0_stride = 128
  g1[6] = 0;
  g1[7] = 0;
  i32x4 z4 = {0, 0, 0, 0};
#if __clang_major__ >= 23
  i32x8 z8 = {0, 0, 0, 0, 0, 0, 0, 0};
  __builtin_amdgcn_tensor_load_to_lds(g0, g1, z4, z4, z8, 0);
#else
  __builtin_amdgcn_tensor_load_to_lds(g0, g1, z4, z4, 0);
#endif
}

// -----------------------------------------------------------------------------
// Kernel 1: P = exp(h_t @ W + bias); scatter into alpha / beta / kappa*0.2
// M = B*U = 4096, N = 30 (padded 32), K = 512. f32 WMMA 16x16x4.
// W staged in LDS pair-interleaved: (k,n) -> sW[(k>>1)*96 + n*2 + (k&1)]
// so each B fragment is a single aligned ds_load_b64 (no repack moves).
// -----------------------------------------------------------------------------
#define PSTR 96
__global__ __launch_bounds__(128) void gauss_param_gemm(
    const float* __restrict__ h, const float* __restrict__ W,
    const float* __restrict__ bias,
    float* __restrict__ alpha, float* __restrict__ beta,
    float* __restrict__ kappa)
{
  __shared__ float sW[(H_ / 2) * PSTR];
  const int tid = threadIdx.x;
  for (int idx = tid; idx < H_ * 32; idx += 128) {
    int r = idx >> 5, c = idx & 31;
    float w = (c < 30) ? W[r * 30 + c] : 0.0f;
    sW[(r >> 1) * PSTR + c * 2 + (r & 1)] = w;
  }
  __syncthreads();

  const int wave  = tid >> 5;
  const int lane  = tid & 31;
  const int mtile = blockIdx.x * 4 + wave;
  const int n0    = lane & 15;          // N index (B frag) == M index (A frag)
  const int kb    = (lane >> 4) * 2;    // K pair base per ISA 16x4 layout

  v8f acc0 = {0.f,0.f,0.f,0.f,0.f,0.f,0.f,0.f};
  v8f acc1 = {0.f,0.f,0.f,0.f,0.f,0.f,0.f,0.f};

  const float* hrow = h + (size_t)(mtile * 16 + n0) * H_;
  for (int k = 0; k < H_; k += 4) {
    float2 av = *(const float2*)(hrow + k + kb);
    v2f a; a.x = av.x; a.y = av.y;
    const float* wp = &sW[((k + kb) >> 1) * PSTR];
    float2 b0v = *(const float2*)(wp + n0 * 2);
    float2 b1v = *(const float2*)(wp + 32 + n0 * 2);
    v2f b0; b0.x = b0v.x; b0.y = b0v.y;
    v2f b1; b1.x = b1v.x; b1.y = b1v.y;
    acc0 = __builtin_amdgcn_wmma_f32_16x16x4_f32(false, a, false, b0,
                                                 (short)0, acc0, false, false);
    acc1 = __builtin_amdgcn_wmma_f32_16x16x4_f32(false, a, false, b1,
                                                 (short)0, acc1, false, false);
  }

  // C/D layout: VGPR i -> M = i (+8 for lanes 16..31), N = lane&15
  const int mh = (lane >> 4) * 8;
  const float bias0 = bias[n0];
  const int n1 = 16 + n0;
  const float bias1 = (n1 < 30) ? bias[n1] : 0.0f;
#pragma unroll
  for (int i = 0; i < 8; ++i) {
    int row = mtile * 16 + i + mh;
    float v0 = __expf(acc0[i] + bias0);
    if (n0 < 10) alpha[row * K_ + n0] = v0;
    else         beta[row * K_ + (n0 - 10)] = v0;
    if (n1 < 30) {
      float v1 = __expf(acc1[i] + bias1);
      if (n1 < 20) beta[row * K_ + (n1 - 10)] = v1;
      else         kappa[row * K_ + (n1 - 20)] = v1 * 0.2f;   // *0.2 before cumsum
    }
  }
}

// -----------------------------------------------------------------------------
// Kernel 2: in-place inclusive cumsum of kappa over U (per b,k).
// One block per batch; Hillis-Steele scan in LDS per k (10 scans of 512).
// -----------------------------------------------------------------------------
__global__ __launch_bounds__(512) void gauss_kappa_cumsum(float* __restrict__ kappa)
{
  __shared__ float s[U_];
  const int b = blockIdx.x;
  const int u = threadIdx.x;
  const int base = (b * U_ + u) * K_;
  for (int k = 0; k < K_; ++k) {
    s[u] = kappa[base + k];
    __syncthreads();
    for (int off = 1; off < U_; off <<= 1) {
      float t = (u >= off) ? s[u - off] : 0.0f;
      __syncthreads();
      s[u] += t;
      __syncthreads();
    }
    kappa[base + k] = s[u];
    __syncthreads();
  }
}

// -----------------------------------------------------------------------------
// Kernel 3: fused phi + out = phi @ (ctx * mask).
// Block = 256 threads (8 waves) per (b, 16-row U tile).
// Double-buffered TDM stages ctx chunks into LDS while all waves compute the
// exp-heavy phi tile (the true bottleneck); each wave then WMMA-accumulates
// its 16-wide D tile from LDS. phi never touches global memory.
// -----------------------------------------------------------------------------
__global__ __launch_bounds__(256) void gauss_phi_ctx(
    const float* __restrict__ ctx, const float* __restrict__ mask,
    const float* __restrict__ alpha, const float* __restrict__ beta,
    const float* __restrict__ kappa, float* __restrict__ out)
{
  __shared__ float sCtx[2][TCHUNK * ROWSTR];  // TDM-padded rows (bank-split halves)
  __shared__ float sPhi[16 * 68];             // stride 68 -> conflict-free A reads
  __shared__ float sPar[3 * 16 * K_];

  const int tid     = threadIdx.x;
  const int b       = blockIdx.y;
  const int mtile   = blockIdx.x;
  const int rowbase = b * U_ + mtile * 16;

  for (int idx = tid; idx < 3 * 16 * K_; idx += 256) {
    int which = idx / 160, rem = idx % 160;
    int uu = rem / K_, kk = rem % K_;
    const float* src = (which == 0) ? alpha : (which == 1) ? beta : kappa;
    sPar[idx] = src[(rowbase + uu) * K_ + kk];
  }

  const int lane = tid & 31;
  const int wave = tid >> 5;
  const int d0   = wave * 16;           // this wave's D tile
  const int n0   = lane & 15;           // N index (B) == M index (A)
  const int kb   = (lane >> 4) * 2;     // K pair base

  const float* ctxb  = ctx  + (size_t)b * T_ * D_;
  const float* maskb = mask + b * T_;
  const unsigned int ldsCtx0 = (unsigned int)(uintptr_t)&sCtx[0][0];
  const unsigned int ldsCtx1 = (unsigned int)(uintptr_t)&sCtx[1][0];

  // Kick off the TDM for chunk 0 (wave 0 only; TDM ignores EXEC).
  if (wave == 0) tdm_load_ctx_chunk(ctxb, ldsCtx0);
  __syncthreads();   // sPar visible

  // phi work assignment: u = tid&15, 4 consecutive t per thread
  const int    pu  = tid & 15;
  const int    pt0 = (tid >> 4) * 4;
  const float* pA  = &sPar[pu * K_];
  const float* pB  = &sPar[160 + pu * K_];
  const float* pK  = &sPar[320 + pu * K_];

  v8f acc = {0.f,0.f,0.f,0.f,0.f,0.f,0.f,0.f};
  const int nchunks = T_ / TCHUNK;

  for (int c = 0; c < nchunks; ++c) {
    const int buf = c & 1;
    const int tc  = c * TCHUNK;

    // --- phi tile for chunk c (overlaps the in-flight TDM of chunk c) ---
#pragma unroll
    for (int j = 0; j < 4; ++j) {
      int tl = pt0 + j;
      float tf = (float)(tc + tl);
      float sum = 0.0f;
#pragma unroll
      for (int k = 0; k < K_; ++k) {
        float d = pK[k] - tf;
        sum += pA[k] * __expf(-pB[k] * d * d);
      }
      sPhi[pu * 68 + tl] = sum * maskb[tc + tl];
    }

    // --- wave 0: issue TDM for chunk c+1, then guarantee chunk c landed ---
    if (wave == 0) {
      if (c + 1 < nchunks) {
        tdm_load_ctx_chunk(ctxb + (size_t)(tc + TCHUNK) * D_,
                           (buf ? ldsCtx0 : ldsCtx1));
        __builtin_amdgcn_s_wait_tensorcnt(1);  // in-order: chunk c complete
      } else {
        __builtin_amdgcn_s_wait_tensorcnt(0);
      }
    }
    __syncthreads();

    // --- WMMA: acc += phi[16u x 64t] @ ctx[64t x 16d] from LDS ---
    const float* cbuf = &sCtx[buf][0];
#pragma unroll 4
    for (int tl = 0; tl < TCHUNK; tl += 4) {
      v2f a, bb;
      a.x = sPhi[n0 * 68 + tl + kb];
      a.y = sPhi[n0 * 68 + tl + kb + 1];
      const float* cb = cbuf + (tl + kb) * ROWSTR + d0 + n0;
      bb.x = cb[0];
      bb.y = cb[ROWSTR];
      acc = __builtin_amdgcn_wmma_f32_16x16x4_f32(false, a, false, bb,
                                                  (short)0, acc, false, false);
    }
    __syncthreads();
  }

  const int mh = (lane >> 4) * 8;
  float* outb = out + (size_t)rowbase * D_;
#pragma unroll
  for (int i = 0; i < 8; ++i) {
    outb[(size_t)(i + mh) * D_ + d0 + n0] = acc[i];
  }
}

// -----------------------------------------------------------------------------
extern "C" void kernel_launch(void* const* d_in, const int* in_sizes, int n_in,
                              void* d_out, int out_size, void* d_ws, size_t ws_size,
                              hipStream_t stream) {
  const float* h    = (const float*)d_in[0];   // [B,U,H]
  const float* ctx  = (const float*)d_in[1];   // [B,T,D]
  const float* mask = (const float*)d_in[2];   // [B,T]
  const float* W    = (const float*)d_in[3];   // [H,3K]
  const float* bias = (const float*)d_in[4];   // [3K]
  float* out = (float*)d_out;                  // [B,U,D]

  float* alpha = (float*)d_ws;                 // [B*U,K]
  float* beta  = alpha + B_ * U_ * K_;
  float* kappa = beta  + B_ * U_ * K_;

  gauss_param_gemm<<<dim3((B_ * U_ / 16) / 4), 128, 0, stream>>>(
      h, W, bias, alpha, beta, kappa);
  gauss_kappa_cumsum<<<dim3(B_), 512, 0, stream>>>(kappa);
  gauss_phi_ctx<<<dim3(U_ / 16, B_), 256, 0, stream>>>(
      ctx, mask, alpha, beta, kappa, out);
}